// JoinSegmenter_5832565588128
// MI455X (gfx1250) — compile-verified
//
#include <hip/hip_runtime.h>
#include <hip/hip_bf16.h>
#include <stdint.h>

// JoinSegmenter.logit_segment (eval mean-pool join) for MI455X / gfx1250.
// Bandwidth-bound op (~390MB traffic -> ~17us floor @ 23.3TB/s); no matmul in
// the math, so the CDNA5 path used is async global->LDS streaming + wave32
// block scans, not WMMA.

#define B_ 64
#define T_ 8192
#define C_ 64
#define ROWF 68   // padded floats per LDS tile row: 272B = 16B-aligned, odd*4 banks

// ---------- CDNA5 async global->LDS helpers (builtin first, asm fallback) ----
typedef int v4i __attribute__((vector_size(16)));
typedef __attribute__((address_space(1))) v4i g_v4i;   // global (AS1) int4
typedef __attribute__((address_space(3))) v4i l_v4i;   // LDS    (AS3) int4

__device__ __forceinline__ void async_g2l_b128(const void* gsrc, void* ldst) {
#if __has_builtin(__builtin_amdgcn_global_load_async_to_lds_b128)
  __builtin_amdgcn_global_load_async_to_lds_b128(
      (g_v4i*)(uintptr_t)gsrc,
      (l_v4i*)(uint32_t)(uintptr_t)ldst,   // low 32 bits of flat = LDS offset
      0, 0);
#else
  asm volatile("global_load_async_to_lds_b128 %0, %1, off"
               :: "v"((uint32_t)(uintptr_t)ldst), "v"(gsrc)
               : "memory");
#endif
}

__device__ __forceinline__ void wait_async0() {
#if __has_builtin(__builtin_amdgcn_s_wait_asynccnt)
  __builtin_amdgcn_s_wait_asynccnt(0);
#else
  asm volatile("s_wait_asynccnt 0" ::: "memory");
#endif
}

// ---------- K1: per-frame argmax over C=64 (padding -> -1) -------------------
// 128 frames/block. Stage 128x64 f32 tile into LDS with async b128 copies
// (perfectly coalesced), then each thread argmaxes its own padded LDS row.
__global__ __launch_bounds__(128) void k1_argmax(const float* __restrict__ logits,
                                                 const unsigned char* __restrict__ pmask,
                                                 int* __restrict__ preds) {
  __shared__ float tile[128 * ROWF];           // 34,816 B
  const int tid = threadIdx.x;
  const size_t f0 = (size_t)blockIdx.x * 128;
  const float* gbase = logits + f0 * C_;

  // 128 rows * 16 chunks(16B) = 2048 chunks; 16 iters * 128 lanes
#pragma unroll
  for (int k = 0; k < 16; ++k) {
    const int g = k * 128 + tid;               // linear 16B chunk id (coalesced)
    const int fr = g >> 4, sub = g & 15;
    async_g2l_b128(gbase + (size_t)g * 4, &tile[fr * ROWF + sub * 4]);
  }
  wait_async0();
  __syncthreads();

  const float* row = &tile[tid * ROWF];
  float best = -3.402823466e38f;
  int bi = 0;
#pragma unroll
  for (int k = 0; k < 16; ++k) {               // strict '>' keeps first max (jnp.argmax)
    float4 v = *(const float4*)(row + 4 * k);
    if (v.x > best) { best = v.x; bi = 4 * k; }
    if (v.y > best) { best = v.y; bi = 4 * k + 1; }
    if (v.z > best) { best = v.z; bi = 4 * k + 2; }
    if (v.w > best) { best = v.w; bi = 4 * k + 3; }
  }
  const size_t fr = f0 + tid;
  preds[fr] = pmask[fr] ? -1 : bi;
}

// ---------- K2: per-row RLE + keep/compact scans -----------------------------
// One 1024-thread block per batch row; 8 frames/thread; two Hillis-Steele scans.
__global__ __launch_bounds__(1024) void k2_scan(const int* __restrict__ preds,
                                                int* __restrict__ seg_start,
                                                int* __restrict__ dstpos,
                                                int* __restrict__ numsegs,
                                                int* __restrict__ nkeep) {
  __shared__ int sc[1024];
  __shared__ int sstart[T_ + 1];               // 32,772 B
  const int b = blockIdx.x, tid = threadIdx.x;
  const int* pr = preds + (size_t)b * T_;
  int* ssg = seg_start + (size_t)b * (T_ + 1);
  int* dpr = dstpos + (size_t)b * T_;

  const int base = tid * 8;
  const int p_prev = (base == 0) ? -100 : pr[base - 1];  // sentinel -> boundary at t=0
  int pv[8];
#pragma unroll
  for (int j = 0; j < 8; ++j) pv[j] = pr[base + j];
  int f[8]; int cnt = 0;
#pragma unroll
  for (int j = 0; j < 8; ++j) {
    f[j] = (pv[j] != ((j == 0) ? p_prev : pv[j - 1])) ? 1 : 0;
    cnt += f[j];
  }

  // scan #1: boundary counts -> segment ids
  sc[tid] = cnt; __syncthreads();
  for (int off = 1; off < 1024; off <<= 1) {
    int v = sc[tid];
    int u = (tid >= off) ? sc[tid - off] : 0;
    __syncthreads();
    sc[tid] = v + u;
    __syncthreads();
  }
  const int incl = sc[tid];
  const int total = sc[1023];                  // num_segs
  int run = incl - cnt;
#pragma unroll
  for (int j = 0; j < 8; ++j) {
    if (f[j]) { sstart[run] = base + j; ssg[run] = base + j; ++run; }
  }
  __syncthreads();
  if (tid == 0) { sstart[total] = T_; ssg[total] = T_; numsegs[b] = total; }
  __syncthreads();

  // scan #2: keep (run value != -1) -> compacted positions
  int kf[8]; int kcnt = 0;
#pragma unroll
  for (int j = 0; j < 8; ++j) {
    const int s = base + j; int k = 0;
    if (s < total) { k = (pr[sstart[s]] != -1) ? 1 : 0; }
    kf[j] = k; kcnt += k;
  }
  sc[tid] = kcnt; __syncthreads();
  for (int off = 1; off < 1024; off <<= 1) {
    int v = sc[tid];
    int u = (tid >= off) ? sc[tid - off] : 0;
    __syncthreads();
    sc[tid] = v + u;
    __syncthreads();
  }
  int run2 = sc[tid] - kcnt;
#pragma unroll
  for (int j = 0; j < 8; ++j) {
    const int s = base + j;
    if (s < total) { dpr[s] = kf[j] ? run2 : -1; run2 += kf[j]; }
  }
  if (tid == 1023) nkeep[b] = sc[1023];
}

// ---------- K3: segment means -> compacted output slots ----------------------
// 64 threads per segment (one per channel) => wave-uniform loop bounds,
// 256B coalesced reads per frame step.
__global__ __launch_bounds__(256) void k3_segmean(const float* __restrict__ logits,
                                                  const int* __restrict__ seg_start,
                                                  const int* __restrict__ dstpos,
                                                  const int* __restrict__ numsegs,
                                                  float* __restrict__ out_logits) {
  const int b = blockIdx.y;
  const int s = blockIdx.x * 4 + (threadIdx.x >> 6);
  const int c = threadIdx.x & 63;
  const int ns = numsegs[b];
  if (s >= ns) return;
  const int dst = dstpos[(size_t)b * T_ + s];
  if (dst < 0) return;                          // dropped padding run
  const int* ss = seg_start + (size_t)b * (T_ + 1);
  const int st = ss[s], en = ss[s + 1];
  const float* src = logits + ((size_t)b * T_ + st) * C_ + c;
  float acc = 0.f;
  for (int t = st; t < en; ++t, src += C_) acc += *src;
  out_logits[((size_t)b * T_ + dst) * C_ + c] = acc / (float)(en - st);
}

// ---------- K4: new_pad + zero-fill dropped/tail slots -----------------------
__global__ __launch_bounds__(256) void k4_pad(const int* __restrict__ nkeep,
                                              float* __restrict__ out_logits,
                                              float* __restrict__ out_pad) {
  const int i = blockIdx.x * 256 + threadIdx.x;  // [0, B*T)
  const int b = i >> 13;                          // T = 8192
  const int j = i & (T_ - 1);
  const int nk = nkeep[b];
  out_pad[i] = (j >= nk) ? 1.0f : 0.0f;
  if (j >= nk) {
    const float4 z = make_float4(0.f, 0.f, 0.f, 0.f);
    float4* p = (float4*)(out_logits + (size_t)i * C_);
#pragma unroll
    for (int k = 0; k < 16; ++k) p[k] = z;
  }
}

extern "C" void kernel_launch(void* const* d_in, const int* in_sizes, int n_in,
                              void* d_out, int out_size, void* d_ws, size_t ws_size,
                              hipStream_t stream) {
  const float* logits = (const float*)d_in[0];
  const unsigned char* pmask = (const unsigned char*)d_in[1];  // jax bool: 1 byte/elem

  float* out_logits = (float*)d_out;                           // [B,T,C]
  float* out_pad = (float*)d_out + (size_t)B_ * T_ * C_;       // [B,T] as 0/1 floats

  // workspace: preds[B*T] | seg_start[B*(T+1)] | dstpos[B*T] | numsegs[B] | nkeep[B]
  int* preds = (int*)d_ws;
  int* seg_start = preds + (size_t)B_ * T_;
  int* dstpos = seg_start + (size_t)B_ * (T_ + 1);
  int* numsegs = dstpos + (size_t)B_ * T_;
  int* nkeep = numsegs + B_;

  k1_argmax<<<dim3((B_ * T_) / 128), 128, 0, stream>>>(logits, pmask, preds);
  k2_scan<<<dim3(B_), 1024, 0, stream>>>(preds, seg_start, dstpos, numsegs, nkeep);
  k3_segmean<<<dim3(T_ / 4, B_), 256, 0, stream>>>(logits, seg_start, dstpos, numsegs, out_logits);
  k4_pad<<<dim3((B_ * T_) / 256), 256, 0, stream>>>(nkeep, out_logits, out_pad);
}